// Model_89902255440605
// MI455X (gfx1250) — compile-verified
//
#include <hip/hip_runtime.h>

typedef __attribute__((ext_vector_type(2))) float v2f;
typedef __attribute__((ext_vector_type(4))) float v4f;
typedef __attribute__((ext_vector_type(8))) float v8f;

#define THREADS 256

__device__ __forceinline__ void atomicAddF32(float* p, float v) {
  // no-return global f32 atomic add (STOREcnt-tracked; s_endpgm waits idle)
  asm volatile("global_atomic_add_f32 %0, %1, off" : : "v"(p), "v"(v) : "memory");
}

// ---------------- degree / normalization ----------------

__global__ void k_deg_init(unsigned* __restrict__ deg, int n) {
  int i = blockIdx.x * blockDim.x + threadIdx.x;
  if (i < n) deg[i] = 1u;  // self-loop contributes 1
}

__global__ void k_deg_count(const long long* __restrict__ eidx,
                            unsigned* __restrict__ deg, int E) {
  int e = blockIdx.x * blockDim.x + threadIdx.x;
  if (e < E) {
    int d = (int)__builtin_nontemporal_load(&eidx[(long long)E + e]);  // NT: don't pollute L2
    atomicAdd(&deg[d], 1u);
  }
}

__global__ void k_deg_fin(unsigned* __restrict__ deg_dinv, int n) {
  int i = blockIdx.x * blockDim.x + threadIdx.x;
  if (i < n) {
    float f = (float)deg_dinv[i];        // deg >= 1 always (self-loop)
    ((float*)deg_dinv)[i] = rsqrtf(f);   // in-place u32 -> f32
  }
}

// ---------------- dense transform via V_WMMA_F32_16X16X4_F32 ----------------
// Full 16-node tiles only — zero predication in the loop.
// xw4[node][0..2] = relu?(in[node]) @ W, xw4[node][3] = 0   (stride-4 padded)
// A (16x4 f32): lane L holds A[M=L%16][k0], A[M][k0+1], k0 = 2*(L/16)
// B (4x16 f32): lane L holds B[k0][N=L%16], B[k0+1][N]
// D (16x16 f32): lane L, VGPR r -> D[M = r + 8*(L/16)][N = L%16]

template <int CIN, bool RELU>
__global__ void k_transform(const float* __restrict__ in, const float* __restrict__ Wp,
                            float* __restrict__ xw4, int ntiles) {
  int tid  = blockIdx.x * blockDim.x + threadIdx.x;
  int lane = threadIdx.x & 31;
  int wave = tid >> 5;
  int nwav = (gridDim.x * blockDim.x) >> 5;
  int col  = lane & 15;   // = N for B/D, = M for A
  int half = lane >> 4;
  int k0   = 2 * half;

  // B: W is [CIN x 3] row-major, padded to 4x16 with zeros
  float wb0 = 0.f, wb1 = 0.f;
  if (col < 3) {
    if (k0 < CIN)     wb0 = Wp[k0 * 3 + col];
    if (k0 + 1 < CIN) wb1 = Wp[(k0 + 1) * 3 + col];
  }
  v2f B = {wb0, wb1};

  for (int t = wave; t < ntiles; t += nwav) {
    int base = t << 4;
    int row  = base + col;                 // A-row node index for this lane
    v2f A;
    if (CIN == 1) {
      float v = in[row];                   // unconditional load (full tile)
      A.x = (half == 0) ? v : 0.f;         // branchless v_cndmask
      A.y = 0.f;
    } else {
      A = ((const v2f*)in)[row * 2 + half];  // h4 padded: aligned b64 load
    }
    if (RELU) { A.x = fmaxf(A.x, 0.f); A.y = fmaxf(A.y, 0.f); }

    v8f C = {};
    C = __builtin_amdgcn_wmma_f32_16x16x4_f32(false, A, false, B,
                                              (short)0, C, false, false);
    int nodeBase = base + half * 8;
    if (col < 4) {                         // col 3 stores the 0 padding for free
#pragma unroll
      for (int r = 0; r < 8; ++r) xw4[(nodeBase + r) * 4 + col] = C[r];
    }
  }
}

// scalar tail for the last (n & 15) nodes — never launched when 16 | N
__global__ void k_transform_tail(const float* __restrict__ in, const float* __restrict__ Wp,
                                 float* __restrict__ xw4, int n, int start,
                                 int cin, int relu_in) {
  int i = start + threadIdx.x;
  if (i < n) {
    float acc0 = 0.f, acc1 = 0.f, acc2 = 0.f;
    for (int k = 0; k < cin; ++k) {
      float v = (cin == 1) ? in[i] : in[i * 4 + k];
      if (relu_in) v = fmaxf(v, 0.f);
      acc0 += v * Wp[k * 3 + 0];
      acc1 += v * Wp[k * 3 + 1];
      acc2 += v * Wp[k * 3 + 2];
    }
    xw4[i * 4 + 0] = acc0; xw4[i * 4 + 1] = acc1;
    xw4[i * 4 + 2] = acc2; xw4[i * 4 + 3] = 0.f;
  }
}

// ---------------- aggregation ----------------

__global__ void k_agg_init(const float* __restrict__ xw4, const float* __restrict__ dinv,
                           const float* __restrict__ bias, float* __restrict__ out4, int n) {
  int i = blockIdx.x * blockDim.x + threadIdx.x;
  if (i < n) {
    float w = dinv[i] * dinv[i];  // self-loop norm
    v4f xv = ((const v4f*)xw4)[i];
    v4f o;
    o.x = xv.x * w + bias[0];
    o.y = xv.y * w + bias[1];
    o.z = xv.z * w + bias[2];
    o.w = 0.f;
    ((v4f*)out4)[i] = o;          // single b128 store
  }
}

__global__ void k_agg_edges(const long long* __restrict__ eidx,
                            const float* __restrict__ xw4, const float* __restrict__ dinv,
                            float* __restrict__ out4, int E) {
  int e = blockIdx.x * blockDim.x + threadIdx.x;
  if (e < E) {
    // NT loads: stream the 256MB index array past L2, keep gather arrays resident
    int s = (int)__builtin_nontemporal_load(&eidx[e]);
    int d = (int)__builtin_nontemporal_load(&eidx[(long long)E + e]);
    float w = dinv[s] * dinv[d];
    v4f xv = ((const v4f*)xw4)[s];         // single aligned b128 gather (L2-resident)
    float* o = (float*)out4 + (size_t)d * 4;
    atomicAddF32(o + 0, xv.x * w);
    atomicAddF32(o + 1, xv.y * w);
    atomicAddF32(o + 2, xv.z * w);
  }
}

// padded [N,4] -> packed [N,3] with final ReLU
__global__ void k_repack_relu(const float* __restrict__ out4, float* __restrict__ out, int n) {
  int i = blockIdx.x * blockDim.x + threadIdx.x;
  if (i < n) {
    v4f v = ((const v4f*)out4)[i];
    out[i * 3 + 0] = fmaxf(v.x, 0.f);
    out[i * 3 + 1] = fmaxf(v.y, 0.f);
    out[i * 3 + 2] = fmaxf(v.z, 0.f);
  }
}

// ---------------- launch ----------------

extern "C" void kernel_launch(void* const* d_in, const int* in_sizes, int n_in,
                              void* d_out, int out_size, void* d_ws, size_t ws_size,
                              hipStream_t stream) {
  const float*     x    = (const float*)d_in[0];
  const long long* eidx = (const long long*)d_in[1];
  const float*     W1   = (const float*)d_in[2];
  const float*     b1   = (const float*)d_in[3];
  const float*     W2   = (const float*)d_in[4];
  const float*     b2   = (const float*)d_in[5];
  const float*     W3   = (const float*)d_in[6];
  const float*     b3   = (const float*)d_in[7];
  float* out = (float*)d_out;

  const int N = in_sizes[0];       // x is [N,1]
  const int E = in_sizes[1] / 2;   // edge_index is [2,E]

  // Workspace: dinv (N f32) | xw4 (N x 4 f32) | h4 (N x 4 f32)  ~ 36 MB
  char* ws = (char*)d_ws;
  size_t off = 0;
  unsigned* deg  = (unsigned*)(ws + off);
  float*    dinv = (float*)deg;
  off += ((size_t)N * 4 + 255) & ~(size_t)255;
  float* xw4 = (float*)(ws + off);
  off += ((size_t)N * 16 + 255) & ~(size_t)255;
  float* h4  = (float*)(ws + off);

  const int gN     = (N + THREADS - 1) / THREADS;
  const int gE     = (E + THREADS - 1) / THREADS;
  const int gT     = 1024;       // 8192 waves grid-striding full 16-node tiles
  const int ntiles = N >> 4;
  const int tail   = N & 15;
  const int tbase  = N & ~15;

  // degrees + symmetric normalization
  k_deg_init <<<gN, THREADS, 0, stream>>>(deg, N);
  k_deg_count<<<gE, THREADS, 0, stream>>>(eidx, deg, E);
  k_deg_fin  <<<gN, THREADS, 0, stream>>>(deg, N);

  // Layer 1: x [N,1] @ W1 [1,3]
  if (ntiles) k_transform<1, false><<<gT, THREADS, 0, stream>>>(x, W1, xw4, ntiles);
  if (tail)   k_transform_tail<<<1, 16, 0, stream>>>(x, W1, xw4, N, tbase, 1, 0);
  k_agg_init <<<gN, THREADS, 0, stream>>>(xw4, dinv, b1, h4, N);
  k_agg_edges<<<gE, THREADS, 0, stream>>>(eidx, xw4, dinv, h4, E);

  // Layer 2: relu(h) @ W2
  if (ntiles) k_transform<3, true><<<gT, THREADS, 0, stream>>>(h4, W2, xw4, ntiles);
  if (tail)   k_transform_tail<<<1, 16, 0, stream>>>(h4, W2, xw4, N, tbase, 3, 1);
  k_agg_init <<<gN, THREADS, 0, stream>>>(xw4, dinv, b2, h4, N);
  k_agg_edges<<<gE, THREADS, 0, stream>>>(eidx, xw4, dinv, h4, E);

  // Layer 3: relu(h) @ W3 -> h4 (reused), then repack+relu into d_out
  if (ntiles) k_transform<3, true><<<gT, THREADS, 0, stream>>>(h4, W3, xw4, ntiles);
  if (tail)   k_transform_tail<<<1, 16, 0, stream>>>(h4, W3, xw4, N, tbase, 3, 1);
  k_agg_init <<<gN, THREADS, 0, stream>>>(xw4, dinv, b3, h4, N);
  k_agg_edges<<<gE, THREADS, 0, stream>>>(eidx, xw4, dinv, h4, E);
  k_repack_relu<<<gN, THREADS, 0, stream>>>(h4, out, N);
}